// EGNN_52226802320141
// MI455X (gfx1250) — compile-verified
//
#include <hip/hip_runtime.h>
#include <math.h>
#include <stdint.h>

// ---------------------------------------------------------------------------
// EGNN forward for MI455X (gfx1250, wave32, WMMA + TDM + async-LDS).
// fp32 end-to-end using V_WMMA_F32_16X16X4_F32.
//   - edge/coord MLP kernels: persistent workgroups, weights staged in LDS via
//     tensor_load_to_lds (TDM), gathers staged via global_load_async_to_lds_b128
//   - node/emb kernels: LDS-staged activations, weights streamed from L2
// ---------------------------------------------------------------------------

typedef __attribute__((ext_vector_type(2))) float v2f;
typedef __attribute__((ext_vector_type(8))) float v8f;
typedef unsigned int v4u __attribute__((ext_vector_type(4)));
typedef int v8i_t __attribute__((ext_vector_type(8)));
typedef int v4i_t __attribute__((ext_vector_type(4)));

static constexpr int HD = 128;   // hidden dim
static constexpr int TE = 64;    // edges / nodes per workgroup tile
static constexpr int PGRID = 512;// persistent grid size for edge/coord kernels

__device__ __forceinline__ float silu_f(float v) {
    return v / (1.0f + __expf(-v));
}

__device__ __forceinline__ void atomic_add_f32(float* p, float v) {
    __hip_atomic_fetch_add(p, v, __ATOMIC_RELAXED, __HIP_MEMORY_SCOPE_AGENT);
}

// --- CDNA5 async global->LDS copy (per-lane 16B), tracked by ASYNCcnt -------
__device__ __forceinline__ void async_copy16(void* lds_dst, const void* gsrc) {
    asm volatile("global_load_async_to_lds_b128 %0, %1, off"
                 :: "v"((unsigned)(uintptr_t)lds_dst), "v"(gsrc)
                 : "memory");
}
__device__ __forceinline__ void wait_async0() {
    asm volatile("s_wait_asynccnt 0x0" ::: "memory");
}

// --- CDNA5 Tensor Data Mover: 2-D fp32 tile (rows x 128) global -> LDS ------
// D# packing per CDNA5 ISA (count=1, data_size=4B,
// tile == whole tensor, type=2). Issue from ONE wave only (TDM ignores EXEC).
// clang-23 / amdgpu-toolchain: 6-arg builtin (g0 v4u, g1 v8i, v4i, v4i, v8i, cpol)
__device__ __forceinline__ void tdm_load_2d(unsigned lds_off, const void* gptr,
                                            unsigned rows, unsigned cols) {
#if __has_builtin(__builtin_amdgcn_tensor_load_to_lds)
    const unsigned long long ga = (unsigned long long)(uintptr_t)gptr;
    v4u g0;
    g0.x = 1u;                                   // count=1, user descriptor
    g0.y = lds_off;                              // lds_addr (bytes)
    g0.z = (unsigned)(ga & 0xffffffffu);         // global_addr[31:0]
    g0.w = (unsigned)((ga >> 32) & 0x01ffffffu)  // global_addr[56:32]
           | 0x80000000u;                        // type=2 (bits 127:126)
    unsigned q0 = (2u << 16);                          // data_size=2 -> 4B
    unsigned q1 = (cols & 0xffffu) << 16;              // tensor_dim0[15:0]
    unsigned q2 = ((cols >> 16) & 0xffffu)             // tensor_dim0[31:16]
                | ((rows & 0xffffu) << 16);            // tensor_dim1[15:0]
    unsigned q3 = ((rows >> 16) & 0xffffu)             // tensor_dim1[31:16]
                | ((cols & 0xffffu) << 16);            // tile_dim0
    unsigned q4 = (rows & 0xffffu);                    // tile_dim1 (tile_dim2=0)
    unsigned q5 = cols;                                // tensor_dim0_stride[31:0]
    v8i_t g1 = {(int)q0, (int)q1, (int)q2, (int)q3, (int)q4, (int)q5, 0, 0};
    v4i_t gz = {0, 0, 0, 0};
    v8i_t gz8 = {0, 0, 0, 0, 0, 0, 0, 0};
    __builtin_amdgcn_tensor_load_to_lds(g0, g1, gz, gz, gz8, 0);
#else
    (void)lds_off; (void)gptr; (void)rows; (void)cols;
#endif
}

__device__ __forceinline__ void wait_tensor0() {
#if __has_builtin(__builtin_amdgcn_s_wait_tensorcnt)
    __builtin_amdgcn_s_wait_tensorcnt(0);
#else
    asm volatile("s_wait_tensorcnt 0x0" ::: "memory");
#endif
}

// Stage W1 (rows1 x 128) and W2 (128 x 128) into LDS, all threads participate.
__device__ __forceinline__ void stage_weights(float* sW1, const float* W1, int rows1,
                                              float* sW2, const float* W2, int tid) {
#if __has_builtin(__builtin_amdgcn_tensor_load_to_lds)
    if (tid < 32) {   // scalar branch: only wave 0 issues TDM ops
        tdm_load_2d((unsigned)(uintptr_t)(void*)sW1, W1, (unsigned)rows1, HD);
        tdm_load_2d((unsigned)(uintptr_t)(void*)sW2, W2, HD, HD);
        wait_tensor0();
    }
#else
    const int n1 = rows1 * HD / 4, n2 = HD * HD / 4;
    for (int i = tid; i < n1; i += 256) ((float4*)sW1)[i] = ((const float4*)W1)[i];
    for (int i = tid; i < n2; i += 256) ((float4*)sW2)[i] = ((const float4*)W2)[i];
#endif
    __syncthreads();
}

// One wave computes MT 16x16 output tiles (output cols n0..n0+15, rows
// rowBase + mt*16 ..), K-loop in steps of 4 with V_WMMA_F32_16X16X4_F32.
// A from LDS; B (weights) from LDS or global, [K x ldb] row-major; B fragment
// loaded once per K-step and reused across all MT M-tiles.
template <int K, int MT>
__device__ __forceinline__ void gemm_tiles(const float* __restrict__ sA, int lda,
                                           int rowBase,
                                           const float* __restrict__ W, int ldb,
                                           int n0, int lane, v8f* acc) {
    const int half = lane >> 4;     // 0 or 1
    const int l    = lane & 15;
    for (int k = 0; k < K; k += 4) {
        const int kk = k + 2 * half;
        v2f b;
        b.x = W[(size_t)kk * ldb + n0 + l];
        b.y = W[(size_t)(kk + 1) * ldb + n0 + l];
#pragma unroll
        for (int mt = 0; mt < MT; ++mt) {
            const float* ap = sA + (size_t)(rowBase + mt * 16 + l) * lda + kk;
            v2f a = *(const v2f*)ap;   // 8B-aligned
            acc[mt] = __builtin_amdgcn_wmma_f32_16x16x4_f32(
                false, a, false, b, (short)0, acc[mt], false, false);
        }
    }
}

// ---------------------------------------------------------------------------
// radial / coord2diff : d[e] = |x[r]-x[c]|^2 ; cdiff = diff/(sqrt(d+1e-8)+1)
// ---------------------------------------------------------------------------
__global__ void radial_kernel(const float* __restrict__ x,
                              const int* __restrict__ ei, int E_,
                              float* __restrict__ d, float* __restrict__ cdiff) {
    const int e = blockIdx.x * 256 + threadIdx.x;
    if (e >= E_) return;
    const int r = ei[e], c = ei[E_ + e];
    const float dx = x[r * 3 + 0] - x[c * 3 + 0];
    const float dy = x[r * 3 + 1] - x[c * 3 + 1];
    const float dz = x[r * 3 + 2] - x[c * 3 + 2];
    const float rad = dx * dx + dy * dy + dz * dz;
    d[e] = rad;
    if (cdiff) {
        const float inv = 1.0f / (sqrtf(rad + 1e-8f) + 1.0f);
        cdiff[e * 3 + 0] = dx * inv;
        cdiff[e * 3 + 1] = dy * inv;
        cdiff[e * 3 + 2] = dz * inv;
    }
}

// ---------------------------------------------------------------------------
// emb: h128 = h16 @ W[16,128] + b      (64 nodes / block, 8 waves)
// ---------------------------------------------------------------------------
__global__ __launch_bounds__(256) void emb_kernel(
    const float* __restrict__ h16, int N_, const float* __restrict__ W,
    const float* __restrict__ b, float* __restrict__ h) {
    __shared__ __align__(16) float sIn[TE][20];
    const int tid = threadIdx.x, v0 = blockIdx.x * TE;
    {
        const int e = tid >> 2, q = tid & 3;
        const int gv = v0 + e;
        float4 val = make_float4(0.f, 0.f, 0.f, 0.f);
        if (gv < N_) val = ((const float4*)(h16 + (size_t)gv * 16))[q];
        *(float4*)&sIn[e][q * 4] = val;
    }
    __syncthreads();
    const int lane = tid & 31, n0 = (tid >> 5) * 16, l = lane & 15, hf = lane >> 4;
    v8f acc[4] = {};
    gemm_tiles<16, 4>(&sIn[0][0], 20, 0, W, HD, n0, lane, acc);
    const float bb = b[n0 + l];
#pragma unroll
    for (int mt = 0; mt < 4; ++mt)
#pragma unroll
        for (int i = 0; i < 8; ++i) {
            const int r = mt * 16 + (hf ? i + 8 : i);
            const int gv = v0 + r;
            if (gv < N_) h[(size_t)gv * HD + n0 + l] = acc[mt][i] + bb;
        }
}

// ---------------------------------------------------------------------------
// Common gather stage for edge-tile kernels: async-copy h[row]|h[col] panels.
// ---------------------------------------------------------------------------
__device__ __forceinline__ void stage_edges(const float* __restrict__ h,
                                            const int* __restrict__ ei, int E_,
                                            const float* __restrict__ d,
                                            const float* __restrict__ d0,
                                            int e0, int tid,
                                            float (*sIn)[260], float (*sEa)[2],
                                            int* sRow) {
    const int e = tid >> 2, q = tid & 3;
    const int ge = e0 + e;
    float* dstp = &sIn[e][q * 64];
    if (ge < E_) {
        const int rr = ei[ge], cc = ei[E_ + ge];
        if (q == 0) { sRow[e] = rr; sEa[e][0] = d[ge]; sEa[e][1] = d0[ge]; }
        const float* src = (q < 2) ? (h + (size_t)rr * HD + q * 64)
                                   : (h + (size_t)cc * HD + (q - 2) * 64);
#pragma unroll
        for (int i = 0; i < 16; ++i) async_copy16(dstp + i * 4, src + i * 4);
    } else {
        if (q == 0) { sRow[e] = -1; sEa[e][0] = 0.f; sEa[e][1] = 0.f; }
        const float4 z = make_float4(0.f, 0.f, 0.f, 0.f);
#pragma unroll
        for (int i = 0; i < 16; ++i) ((float4*)dstp)[i] = z;
    }
    wait_async0();   // drain this wave's async copies before the barrier
}

// ---------------------------------------------------------------------------
// edge MLP (gcl message): m = silu(W2(silu(W1 [h_r|h_c|d|d0] + b1)) + b2)
// then agg[row] += m / 100. Persistent WGs; W1/W2 LDS-resident via TDM.
// ---------------------------------------------------------------------------
__global__ __launch_bounds__(256) void edge_mlp_kernel(
    const float* __restrict__ h, const int* __restrict__ ei, int E_, int nTiles,
    const float* __restrict__ W1, const float* __restrict__ b1,
    const float* __restrict__ W2, const float* __restrict__ b2,
    const float* __restrict__ d, const float* __restrict__ d0,
    float* __restrict__ agg) {
    __shared__ __align__(16) float sW1[258 * HD];   // 132 KB
    __shared__ __align__(16) float sW2[HD * HD];    //  64 KB
    __shared__ __align__(16) float sIn[TE][260];    //  66.5 KB
    __shared__ __align__(16) float sHid[TE][132];   //  33.8 KB
    __shared__ float sEa[TE][2];
    __shared__ int sRow[TE];
    const int tid = threadIdx.x;
    stage_weights(sW1, W1, 258, sW2, W2, tid);

    const int lane = tid & 31, n0 = (tid >> 5) * 16, l = lane & 15, hf = lane >> 4;
    const float bb1 = b1[n0 + l];
    const float wd  = W1[(size_t)256 * HD + n0 + l];
    const float wd0 = W1[(size_t)257 * HD + n0 + l];
    const float bb2 = b2[n0 + l];

    for (int tile = blockIdx.x; tile < nTiles; tile += gridDim.x) {
        const int e0 = tile * TE;
        stage_edges(h, ei, E_, d, d0, e0, tid, sIn, sEa, sRow);
        __syncthreads();
        v8f acc[4] = {};
        gemm_tiles<256, 4>(&sIn[0][0], 260, 0, sW1, HD, n0, lane, acc);
#pragma unroll
        for (int mt = 0; mt < 4; ++mt)
#pragma unroll
            for (int i = 0; i < 8; ++i) {
                const int r = mt * 16 + (hf ? i + 8 : i);
                const float v = acc[mt][i] + bb1 + sEa[r][0] * wd + sEa[r][1] * wd0;
                sHid[r][n0 + l] = silu_f(v);
            }
        __syncthreads();
        v8f acc2[4] = {};
        gemm_tiles<128, 4>(&sHid[0][0], 132, 0, sW2, HD, n0, lane, acc2);
#pragma unroll
        for (int mt = 0; mt < 4; ++mt)
#pragma unroll
            for (int i = 0; i < 8; ++i) {
                const int r = mt * 16 + (hf ? i + 8 : i);
                const int dstRow = sRow[r];
                if (dstRow >= 0) {
                    const float v = silu_f(acc2[mt][i] + bb2);
                    atomic_add_f32(agg + (size_t)dstRow * HD + n0 + l, v * 0.01f);
                }
            }
        __syncthreads();   // protect sIn/sHid before next tile restages
    }
}

// ---------------------------------------------------------------------------
// node MLP: h = h + W2(silu(W1 [h|agg] + b1)) + b2
// ---------------------------------------------------------------------------
__global__ __launch_bounds__(256) void node_mlp_kernel(
    float* __restrict__ h, const float* __restrict__ agg, int N_,
    const float* __restrict__ W1, const float* __restrict__ b1,
    const float* __restrict__ W2, const float* __restrict__ b2) {
    __shared__ __align__(16) float sIn[TE][260];
    __shared__ __align__(16) float sHid[TE][132];
    const int tid = threadIdx.x;
    const int v0 = blockIdx.x * TE;
    {
        const int e = tid >> 2, q = tid & 3;
        const int gv = v0 + e;
        float* dstp = &sIn[e][q * 64];
        if (gv < N_) {
            const float* src = (q < 2) ? (h + (size_t)gv * HD + q * 64)
                                       : (agg + (size_t)gv * HD + (q - 2) * 64);
#pragma unroll
            for (int i = 0; i < 16; ++i) async_copy16(dstp + i * 4, src + i * 4);
        } else {
            const float4 z = make_float4(0.f, 0.f, 0.f, 0.f);
#pragma unroll
            for (int i = 0; i < 16; ++i) ((float4*)dstp)[i] = z;
        }
        wait_async0();
    }
    __syncthreads();
    const int lane = tid & 31, n0 = (tid >> 5) * 16, l = lane & 15, hf = lane >> 4;
    v8f acc[4] = {};
    gemm_tiles<256, 4>(&sIn[0][0], 260, 0, W1, HD, n0, lane, acc);
    {
        const float bb = b1[n0 + l];
#pragma unroll
        for (int mt = 0; mt < 4; ++mt)
#pragma unroll
            for (int i = 0; i < 8; ++i) {
                const int r = mt * 16 + (hf ? i + 8 : i);
                sHid[r][n0 + l] = silu_f(acc[mt][i] + bb);
            }
    }
    __syncthreads();
    v8f acc2[4] = {};
    gemm_tiles<128, 4>(&sHid[0][0], 132, 0, W2, HD, n0, lane, acc2);
    {
        const float bb = b2[n0 + l];
#pragma unroll
        for (int mt = 0; mt < 4; ++mt)
#pragma unroll
            for (int i = 0; i < 8; ++i) {
                const int r = mt * 16 + (hf ? i + 8 : i);
                const int gv = v0 + r;
                if (gv < N_)   // residual: old h lives in sIn[:, 0:128]
                    h[(size_t)gv * HD + n0 + l] = sIn[r][n0 + l] + acc2[mt][i] + bb;
            }
    }
}

// ---------------------------------------------------------------------------
// coord MLP: phi = silu(c2(silu(c1 inp))) @ c3w ; x[row] += cdiff*phi/100
// Persistent WGs; weights LDS-resident via TDM.
// ---------------------------------------------------------------------------
__global__ __launch_bounds__(256) void coord_mlp_kernel(
    const float* __restrict__ h, const int* __restrict__ ei, int E_, int nTiles,
    const float* __restrict__ W1, const float* __restrict__ b1,
    const float* __restrict__ W2, const float* __restrict__ b2,
    const float* __restrict__ c3, const float* __restrict__ d,
    const float* __restrict__ d0, const float* __restrict__ cdiff,
    float* __restrict__ x) {
    __shared__ __align__(16) float sW1[258 * HD];
    __shared__ __align__(16) float sW2[HD * HD];
    __shared__ __align__(16) float sIn[TE][260];
    __shared__ __align__(16) float sHid[TE][132];
    __shared__ float sEa[TE][2];
    __shared__ int sRow[TE];
    const int tid = threadIdx.x;
    stage_weights(sW1, W1, 258, sW2, W2, tid);

    const int lane = tid & 31, n0 = (tid >> 5) * 16, l = lane & 15, hf = lane >> 4;
    const float bb1 = b1[n0 + l];
    const float wd  = W1[(size_t)256 * HD + n0 + l];
    const float wd0 = W1[(size_t)257 * HD + n0 + l];
    const float bb2 = b2[n0 + l];
    const float cw  = c3[n0 + l];

    for (int tile = blockIdx.x; tile < nTiles; tile += gridDim.x) {
        const int e0 = tile * TE;
        stage_edges(h, ei, E_, d, d0, e0, tid, sIn, sEa, sRow);
        __syncthreads();
        v8f acc[4] = {};
        gemm_tiles<256, 4>(&sIn[0][0], 260, 0, sW1, HD, n0, lane, acc);
#pragma unroll
        for (int mt = 0; mt < 4; ++mt)
#pragma unroll
            for (int i = 0; i < 8; ++i) {
                const int r = mt * 16 + (hf ? i + 8 : i);
                const float v = acc[mt][i] + bb1 + sEa[r][0] * wd + sEa[r][1] * wd0;
                sHid[r][n0 + l] = silu_f(v);
            }
        __syncthreads();
        v8f acc2[4] = {};
        gemm_tiles<128, 4>(&sHid[0][0], 132, 0, sW2, HD, n0, lane, acc2);
        __syncthreads();   // all waves done reading sHid before overwrite
#pragma unroll
        for (int mt = 0; mt < 4; ++mt)
#pragma unroll
            for (int i = 0; i < 8; ++i) {
                const int r = mt * 16 + (hf ? i + 8 : i);
                sHid[r][n0 + l] = silu_f(acc2[mt][i] + bb2) * cw;
            }
        __syncthreads();
        if (tid < TE) {
            const int ge = e0 + tid;
            const int rr = sRow[tid];
            if (ge < E_ && rr >= 0) {
                float phi = 0.f;
#pragma unroll 8
                for (int c = 0; c < HD; ++c) phi += sHid[tid][c];
                phi *= 0.01f;   // 'sum' aggregation / NORM_FACTOR
#pragma unroll
                for (int j = 0; j < 3; ++j)
                    atomic_add_f32(x + (size_t)rr * 3 + j,
                                   cdiff[(size_t)ge * 3 + j] * phi);
            }
        }
        __syncthreads();
    }
}

// ---------------------------------------------------------------------------
// emb_out: out16 = h @ W[128,16] + b   (4 waves, each one M-tile)
// ---------------------------------------------------------------------------
__global__ __launch_bounds__(128) void emb_out_kernel(
    const float* __restrict__ h, int N_, const float* __restrict__ W,
    const float* __restrict__ b, float* __restrict__ out) {
    __shared__ __align__(16) float sIn[TE][132];
    const int tid = threadIdx.x, v0 = blockIdx.x * TE;
    {
        const int e = tid >> 1, q = tid & 1;
        const int gv = v0 + e;
        float4* dst = (float4*)&sIn[e][q * 64];
        if (gv < N_) {
            const float* src = h + (size_t)gv * HD + q * 64;
#pragma unroll
            for (int i = 0; i < 16; ++i) dst[i] = ((const float4*)src)[i];
        } else {
            const float4 z = make_float4(0.f, 0.f, 0.f, 0.f);
#pragma unroll
            for (int i = 0; i < 16; ++i) dst[i] = z;
        }
    }
    __syncthreads();
    const int wave = tid >> 5, lane = tid & 31, l = lane & 15, hf = lane >> 4;
    v8f acc = {};
    gemm_tiles<128, 1>(&sIn[0][0], 132, wave * 16, W, 16, 0, lane, &acc);
    const float bb = b[l];
#pragma unroll
    for (int i = 0; i < 8; ++i) {
        const int r = wave * 16 + (hf ? i + 8 : i);
        const int gv = v0 + r;
        if (gv < N_) out[(size_t)gv * 16 + l] = acc[i] + bb;
    }
}

// ---------------------------------------------------------------------------
// Host orchestration. Param flattening (insertion order, DFS):
//  0:h 1:x 2:edge_index 3:emb.w 4:emb.b 5:emb_out.w 6:emb_out.b
//  per block b (base = 7 + 21*b):
//    gcl g (gi = base + 8*g): e1.w e1.b e2.w e2.b n1.w n1.b n2.w n2.b
//    coord (ci = base+16):    c1.w c1.b c2.w c2.b c3w
// ---------------------------------------------------------------------------
extern "C" void kernel_launch(void* const* d_in, const int* in_sizes, int n_in,
                              void* d_out, int out_size, void* d_ws, size_t ws_size,
                              hipStream_t stream) {
    (void)n_in; (void)out_size; (void)ws_size;
    const float* h16  = (const float*)d_in[0];
    const float* x_in = (const float*)d_in[1];
    const int*   ei   = (const int*)d_in[2];
    const int N_ = in_sizes[0] / 16;
    const int E_ = in_sizes[2] / 2;
    auto P = [&](int i) { return (const float*)d_in[i]; };

    float* ws    = (float*)d_ws;
    float* hbuf  = ws;  ws += (size_t)N_ * HD;
    float* agg   = ws;  ws += (size_t)N_ * HD;
    float* xw    = ws;  ws += (size_t)N_ * 3;
    float* d0    = ws;  ws += E_;
    float* dd    = ws;  ws += E_;
    float* cdiff = ws;  ws += (size_t)E_ * 3;

    (void)hipMemcpyAsync(xw, x_in, (size_t)N_ * 3 * sizeof(float),
                         hipMemcpyDeviceToDevice, stream);

    const int nodeBlocks = (N_ + TE - 1) / TE;
    const int edgeTiles  = (E_ + TE - 1) / TE;
    const int pgrid      = edgeTiles < PGRID ? edgeTiles : PGRID;
    const int eltBlocks  = (E_ + 255) / 256;

    radial_kernel<<<eltBlocks, 256, 0, stream>>>(x_in, ei, E_, d0, nullptr);
    emb_kernel<<<nodeBlocks, 256, 0, stream>>>(h16, N_, P(3), P(4), hbuf);

    for (int b = 0; b < 3; ++b) {
        const int base = 7 + 21 * b;
        radial_kernel<<<eltBlocks, 256, 0, stream>>>(xw, ei, E_, dd, cdiff);
        for (int g = 0; g < 2; ++g) {
            const int gi = base + 8 * g;
            (void)hipMemsetAsync(agg, 0, (size_t)N_ * HD * sizeof(float), stream);
            edge_mlp_kernel<<<pgrid, 256, 0, stream>>>(
                hbuf, ei, E_, edgeTiles, P(gi + 0), P(gi + 1), P(gi + 2),
                P(gi + 3), dd, d0, agg);
            node_mlp_kernel<<<nodeBlocks, 256, 0, stream>>>(
                hbuf, agg, N_, P(gi + 4), P(gi + 5), P(gi + 6), P(gi + 7));
        }
        const int ci = base + 16;
        coord_mlp_kernel<<<pgrid, 256, 0, stream>>>(
            hbuf, ei, E_, edgeTiles, P(ci + 0), P(ci + 1), P(ci + 2), P(ci + 3),
            P(ci + 4), dd, d0, cdiff, xw);
    }
    emb_out_kernel<<<nodeBlocks, 128, 0, stream>>>(hbuf, N_, P(5), P(6),
                                                   (float*)d_out);
    (void)hipMemcpyAsync((float*)d_out + (size_t)N_ * 16, xw,
                         (size_t)N_ * 3 * sizeof(float),
                         hipMemcpyDeviceToDevice, stream);
}